// GATEncoder_58643483460295
// MI455X (gfx1250) — compile-verified
//
#include <hip/hip_runtime.h>
#include <stdint.h>

typedef __attribute__((ext_vector_type(2))) float v2f;
typedef __attribute__((ext_vector_type(8))) float v8f;

#define NEG_SLOPE 0.2f

// ---------- helpers ----------
__device__ __forceinline__ unsigned f2key(float f) {
    unsigned u = __float_as_uint(f);
    return (u & 0x80000000u) ? ~u : (u | 0x80000000u);
}
__device__ __forceinline__ float key2f(unsigned k) {
    unsigned u = (k & 0x80000000u) ? (k & 0x7FFFFFFFu) : ~k;
    return __uint_as_float(u);
}
__device__ __forceinline__ float leaky(float x) { return x > 0.0f ? x : NEG_SLOPE * x; }

// ---------- GEMM: C[M,128] = A[M,K] @ B[K,128], f32 WMMA 16x16x4 ----------
// Block: 256 threads = 8 waves; each block owns one 16-row tile, each wave one
// 16-col tile. A tile staged in LDS (16*K floats), B read from global (L2-resident).
template <int K>
__global__ __launch_bounds__(256) void gemm_wmma_f32(
    const float* __restrict__ A, const float* __restrict__ B,
    float* __restrict__ C, int M) {
    __shared__ float As[16 * K];
    const int tid  = threadIdx.x;
    const int tile = blockIdx.x;
    const bool full = (tile * 16 + 16) <= M;   // uniform per block

    // coalesced float4 staging of the 16xK A tile (rows are contiguous)
    const float4* Av  = (const float4*)(A + (size_t)tile * 16 * K);
    float4*       Asv = (float4*)As;
    if (full) {
        for (int i = tid; i < 16 * K / 4; i += 256) Asv[i] = Av[i];
    } else {
        for (int i = tid; i < 16 * K / 4; i += 256) {
            int r = (i * 4) / K;
            float4 v = make_float4(0.f, 0.f, 0.f, 0.f);
            if (tile * 16 + r < M) v = Av[i];
            Asv[i] = v;
        }
    }
    __syncthreads();

    const int lane = tid & 31;
    const int wave = tid >> 5;          // 0..7 -> column tile
    const int cn   = wave * 16;
    const int row  = lane & 15;         // A row within tile
    const int kb   = (lane >> 4) * 2;   // K sub-offset per A/B f32 layout
    const int col  = cn + (lane & 15);  // B/C column

    v8f acc = {};
    for (int k0 = 0; k0 < K; k0 += 4) {
        v2f a, b;
        a.x = As[row * K + k0 + kb];
        a.y = As[row * K + k0 + kb + 1];
        b.x = B[(size_t)(k0 + kb) * 128 + col];
        b.y = B[(size_t)(k0 + kb + 1) * 128 + col];
        acc = __builtin_amdgcn_wmma_f32_16x16x4_f32(
            false, a, false, b, (short)0, acc, false, false);
    }

    // C/D layout: VGPR i -> M = i + (lane>=16 ? 8 : 0), N = lane&15
    const int mbase = (lane >> 4) * 8;
    float* Cp = C + ((size_t)tile * 16 + mbase) * 128 + col;
    if (full) {
#pragma unroll
        for (int i = 0; i < 8; ++i) Cp[(size_t)i * 128] = acc[i];
    } else {
#pragma unroll
        for (int i = 0; i < 8; ++i)
            if (tile * 16 + mbase + i < M) Cp[(size_t)i * 128] = acc[i];
    }
}

// ---------- per-node attention logits + emax init (self-loop) ----------
__global__ void alpha_kernel(const float* __restrict__ h,
                             const float* __restrict__ att_src,
                             const float* __restrict__ att_dst,
                             float* __restrict__ as_, float* __restrict__ ad_,
                             unsigned* __restrict__ emax,
                             int N_, int H, int D) {
    int i = blockIdx.x * blockDim.x + threadIdx.x;
    if (i >= N_ * H) return;
    int n = i / H, hh = i % H;
    const float* hp  = h + (size_t)n * H * D + (size_t)hh * D;
    const float* sp  = att_src + hh * D;
    const float* dp  = att_dst + hh * D;
    float s = 0.f, d = 0.f;
    for (int k = 0; k < D; ++k) { float v = hp[k]; s += v * sp[k]; d += v * dp[k]; }
    as_[i] = s;
    ad_[i] = d;
    emax[i] = f2key(leaky(s + d));   // self-loop edge seeds the segment max
}

// ---------- edge pass 1: segment max via ordered-uint atomicMax ----------
__global__ void edge_max_kernel(const int* __restrict__ src, const int* __restrict__ dst,
                                const float* __restrict__ as_, const float* __restrict__ ad_,
                                unsigned* __restrict__ emax, int E_, int H) {
    int i = blockIdx.x * blockDim.x + threadIdx.x;
    if (i >= E_ * H) return;
    int e = i / H, hh = i % H;
    int s = src[e], d = dst[e];
    float v = leaky(as_[s * H + hh] + ad_[d * H + hh]);
    atomicMax(&emax[d * H + hh], f2key(v));
}

// ---------- node init: self-loop contribution (assignment, no memset needed) ----------
// 128 features per node in both layers (H*D == 128).
__global__ void node_init_kernel(const float* __restrict__ h,
                                 const float* __restrict__ as_, const float* __restrict__ ad_,
                                 const unsigned* __restrict__ emax,
                                 float* __restrict__ denom, float* __restrict__ accum,
                                 int N_, int H) {
    int i = blockIdx.x * blockDim.x + threadIdx.x;
    if (i >= N_ * 128) return;
    int n = i >> 7, f = i & 127;
    int hh = (f * H) >> 7;              // f / (128/H)
    int ih = n * H + hh;
    float e  = leaky(as_[ih] + ad_[ih]);
    float ee = __expf(e - key2f(emax[ih]));
    accum[i] = ee * h[i];
    if ((f & (128 / H - 1)) == 0) denom[ih] = ee;
}

// ---------- edge pass 2: one wave32 per edge, 4 features per lane ----------
__global__ __launch_bounds__(256) void edge_scatter_kernel(
    const int* __restrict__ src, const int* __restrict__ dst,
    const float* __restrict__ as_, const float* __restrict__ ad_,
    const unsigned* __restrict__ emax, const float* __restrict__ h,
    float* __restrict__ denom, float* __restrict__ accum, int E_, int H) {
    int gid  = blockIdx.x * blockDim.x + threadIdx.x;
    int e    = gid >> 5;
    int lane = gid & 31;
    if (e >= E_) return;
    int s = src[e], d = dst[e];
    int hh = (lane * H) >> 5;           // lanes 0-15 head0, 16-31 head1 (H=2); 0 (H=1)
    int ih = d * H + hh;
    float ev = leaky(as_[s * H + hh] + ad_[ih]);
    float ee = __expf(ev - key2f(emax[ih]));
    if ((lane & (32 / H - 1)) == 0) atomicAdd(&denom[ih], ee);
    const float4 hv = *(const float4*)(h + (size_t)s * 128 + lane * 4);
    float* ap = accum + (size_t)d * 128 + lane * 4;
    atomicAdd(ap + 0, ee * hv.x);
    atomicAdd(ap + 1, ee * hv.y);
    atomicAdd(ap + 2, ee * hv.z);
    atomicAdd(ap + 3, ee * hv.w);
}

// ---------- finalize: normalize, bias, optional ELU ----------
__global__ void finalize_kernel(const float* __restrict__ denom,
                                const float* __restrict__ bias,
                                const float* __restrict__ accum,
                                float* __restrict__ outp,
                                int N_, int H, int do_elu) {
    int i = blockIdx.x * blockDim.x + threadIdx.x;
    if (i >= N_ * 128) return;
    int n = i >> 7, f = i & 127;
    int hh = (f * H) >> 7;
    float v = accum[i] / denom[n * H + hh] + bias[f];
    if (do_elu) v = v > 0.f ? v : (__expf(v) - 1.0f);
    outp[i] = v;
}

// ---------- host launcher ----------
extern "C" void kernel_launch(void* const* d_in, const int* in_sizes, int n_in,
                              void* d_out, int out_size, void* d_ws, size_t ws_size,
                              hipStream_t stream) {
    const float* x        = (const float*)d_in[0];
    const int*   edge     = (const int*)d_in[1];
    const float* W1       = (const float*)d_in[2];
    const float* att_src1 = (const float*)d_in[3];
    const float* att_dst1 = (const float*)d_in[4];
    const float* b1       = (const float*)d_in[5];
    const float* W2       = (const float*)d_in[6];
    const float* att_src2 = (const float*)d_in[7];
    const float* att_dst2 = (const float*)d_in[8];
    const float* b2       = (const float*)d_in[9];

    const int N_ = in_sizes[0] / 256;   // IN = 256
    const int E_ = in_sizes[1] / 2;
    const int* src = edge;
    const int* dst = edge + E_;

    // workspace carve-up (256B aligned slabs)
    char* p = (char*)d_ws;
    auto alloc = [&](size_t bytes) -> void* {
        void* r = (void*)p;
        p += (bytes + 255) & ~(size_t)255;
        return r;
    };
    float*    h1     = (float*)alloc((size_t)N_ * 128 * 4);
    float*    accum1 = (float*)alloc((size_t)N_ * 128 * 4);   // becomes ELU(h) in place
    float*    h2     = (float*)alloc((size_t)N_ * 128 * 4);
    float*    accum2 = (float*)alloc((size_t)N_ * 128 * 4);
    float*    as1    = (float*)alloc((size_t)N_ * 2 * 4);
    float*    ad1    = (float*)alloc((size_t)N_ * 2 * 4);
    unsigned* emax1  = (unsigned*)alloc((size_t)N_ * 2 * 4);
    float*    denom1 = (float*)alloc((size_t)N_ * 2 * 4);
    float*    as2    = (float*)alloc((size_t)N_ * 4);
    float*    ad2    = (float*)alloc((size_t)N_ * 4);
    unsigned* emax2  = (unsigned*)alloc((size_t)N_ * 4);
    float*    denom2 = (float*)alloc((size_t)N_ * 4);
    (void)ws_size; (void)n_in; (void)out_size;

    const int rowTiles = (N_ + 15) / 16;
    auto cdiv = [](long long a, long long b) { return (int)((a + b - 1) / b); };

    // ----- layer 1 (H=2, D=64) -----
    gemm_wmma_f32<256><<<rowTiles, 256, 0, stream>>>(x, W1, h1, N_);
    alpha_kernel<<<cdiv((long long)N_ * 2, 256), 256, 0, stream>>>(
        h1, att_src1, att_dst1, as1, ad1, emax1, N_, 2, 64);
    edge_max_kernel<<<cdiv((long long)E_ * 2, 256), 256, 0, stream>>>(
        src, dst, as1, ad1, emax1, E_, 2);
    node_init_kernel<<<cdiv((long long)N_ * 128, 256), 256, 0, stream>>>(
        h1, as1, ad1, emax1, denom1, accum1, N_, 2);
    edge_scatter_kernel<<<cdiv((long long)E_ * 32, 256), 256, 0, stream>>>(
        src, dst, as1, ad1, emax1, h1, denom1, accum1, E_, 2);
    finalize_kernel<<<cdiv((long long)N_ * 128, 256), 256, 0, stream>>>(
        denom1, b1, accum1, accum1, N_, 2, /*elu=*/1);

    // ----- layer 2 (H=1, D=128) -----
    gemm_wmma_f32<128><<<rowTiles, 256, 0, stream>>>(accum1, W2, h2, N_);
    alpha_kernel<<<cdiv((long long)N_, 256), 256, 0, stream>>>(
        h2, att_src2, att_dst2, as2, ad2, emax2, N_, 1, 128);
    edge_max_kernel<<<cdiv((long long)E_, 256), 256, 0, stream>>>(
        src, dst, as2, ad2, emax2, E_, 1);
    node_init_kernel<<<cdiv((long long)N_ * 128, 256), 256, 0, stream>>>(
        h2, as2, ad2, emax2, denom2, accum2, N_, 1);
    edge_scatter_kernel<<<cdiv((long long)E_ * 32, 256), 256, 0, stream>>>(
        src, dst, as2, ad2, emax2, h2, denom2, accum2, E_, 1);
    finalize_kernel<<<cdiv((long long)N_ * 128, 256), 256, 0, stream>>>(
        denom2, b2, accum2, (float*)d_out, N_, 1, /*elu=*/0);
}